// ManifoldPointCloudLayer_15676630630928
// MI455X (gfx1250) — compile-verified
//
#include <hip/hip_runtime.h>

typedef __attribute__((ext_vector_type(2))) float v2f;
typedef __attribute__((ext_vector_type(4))) float v4f;
typedef __attribute__((ext_vector_type(8))) float v8f;

#define N_PTS 4096
#define M_PTS 65536
#define DIM   64
#define WAVES 8
#define MCHUNK (M_PTS / WAVES)   // 8192 P-rows per wave, 512 tiles, 256 loop trips

// Kernel 1: pnorm[m] = sum_d P[m,d]^2
__global__ __launch_bounds__(256) void pnorm_kernel(const float* __restrict__ P,
                                                    float* __restrict__ pn) {
  int m = blockIdx.x * 256 + threadIdx.x;
  const v4f* p4 = (const v4f*)(P + (size_t)m * DIM);
  float s = 0.f;
#pragma unroll
  for (int j = 0; j < DIM / 4; ++j) {
    v4f v = p4[j];
    s += v.x * v.x + v.y * v.y + v.z * v.z + v.w * v.w;
  }
  pn[m] = s;
}

// Kernel 2: per 16 X-rows, argmin over all M via fp32 WMMA tiles, then gather.
__global__ __launch_bounds__(256) void nn_kernel(const float* __restrict__ X,
                                                 const float* __restrict__ P,
                                                 const float* __restrict__ pn,
                                                 float* __restrict__ out) {
  __shared__ float s_val[16 * 128];
  __shared__ int   s_idx[16 * 128];
  __shared__ int   s_best[16];

  const int tid  = threadIdx.x;
  const int wave = tid >> 5;       // 0..7
  const int lane = tid & 31;
  const int half = lane >> 4;      // 0 or 1
  const int l16  = lane & 15;
  const int row0 = blockIdx.x * 16;

  // A operand (X tile 16xK=64), ISA 16x4 f32 A layout per step j (K=4j..4j+3):
  //   lanes 0-15 : (K=4j, 4j+1) of row l16
  //   lanes 16-31: (K=4j+2,4j+3) of row l16
  v2f a[16];
  {
    const float* xrow = X + (size_t)(row0 + l16) * DIM + 2 * half;
#pragma unroll
    for (int j = 0; j < 16; ++j) a[j] = *(const v2f*)(xrow + 4 * j);
  }

  float bestv[8];
  int   besti[8];
#pragma unroll
  for (int v = 0; v < 8; ++v) { bestv[v] = 3.4e38f; besti[v] = 0; }

  const int m_begin = wave * MCHUNK;
  const int m_end   = m_begin + MCHUNK;

  // One 16x16 tile: K=64 as 16 WMMA steps split into two independent
  // accumulator chains (even/odd j) to keep the matrix pipe fed.
  auto process = [&](const v2f* __restrict__ b, int m0v) {
    float pnv = pn[m0v + l16];     // issued early; hides under WMMAs
    v8f acc0 = {};
    v8f acc1 = {};
#pragma unroll
    for (int j = 0; j < 16; j += 2) {
      acc0 = __builtin_amdgcn_wmma_f32_16x16x4_f32(false, a[j],     false, b[j],
                                                   (short)0, acc0, false, false);
      acc1 = __builtin_amdgcn_wmma_f32_16x16x4_f32(false, a[j + 1], false, b[j + 1],
                                                   (short)0, acc1, false, false);
    }
    // C layout: VGPR v -> row M = v + 8*half, col N = l16 (fixed m per lane).
#pragma unroll
    for (int v = 0; v < 8; ++v) {
      float s = pnv - 2.0f * (acc0[v] + acc1[v]);
      if (s < bestv[v]) { bestv[v] = s; besti[v] = m0v + l16; }
    }
  };

  // Register ping-pong double buffering of the B tile (16 x b64 per tile).
  v2f b0[16], b1[16];
  {
    const float* prow = P + (size_t)(m_begin + l16) * DIM + 2 * half;
#pragma unroll
    for (int j = 0; j < 16; ++j) b0[j] = *(const v2f*)(prow + 4 * j);
  }

  for (int m0 = m_begin; m0 < m_end; m0 += 32) {
    {   // prefetch tile m0+16 while computing tile m0
      const float* prow = P + (size_t)(m0 + 16 + l16) * DIM + 2 * half;
#pragma unroll
      for (int j = 0; j < 16; ++j) b1[j] = *(const v2f*)(prow + 4 * j);
    }
    process(b0, m0);

    {   // prefetch tile m0+32 (wrap on last trip; uniform, harmless)
      int mn = (m0 + 32 < m_end) ? (m0 + 32) : m_begin;
      const float* prow = P + (size_t)(mn + l16) * DIM + 2 * half;
#pragma unroll
      for (int j = 0; j < 16; ++j) b0[j] = *(const v2f*)(prow + 4 * j);
    }
    process(b1, m0 + 16);
  }

  // Stage per-lane winners: row r gets 128 candidates (8 waves x 16 cols).
#pragma unroll
  for (int v = 0; v < 8; ++v) {
    int r = v + 8 * half;
    s_val[r * 128 + wave * 16 + l16] = bestv[v];
    s_idx[r * 128 + wave * 16 + l16] = besti[v];
  }
  __syncthreads();

  if (tid < 16) {
    float bv = 3.4e38f;
    int   bi = 0;
    for (int c = 0; c < 128; ++c) {
      float v = s_val[tid * 128 + c];
      if (v < bv) { bv = v; bi = s_idx[tid * 128 + c]; }
    }
    s_best[tid] = bi;
  }
  __syncthreads();

  // Gather winning P rows: 16 rows x 64 floats, one float4 per thread.
  {
    int r = tid >> 4;
    int d = (tid & 15) * 4;
    int m = s_best[r];
    *(v4f*)(out + (size_t)(row0 + r) * DIM + d) =
        *(const v4f*)(P + (size_t)m * DIM + d);
  }
}

extern "C" void kernel_launch(void* const* d_in, const int* in_sizes, int n_in,
                              void* d_out, int out_size, void* d_ws, size_t ws_size,
                              hipStream_t stream) {
  const float* X = (const float*)d_in[0];            // (4096, 64) fp32
  const float* P = (const float*)d_in[1];            // (65536, 64) fp32
  // d_in[2], d_in[3] are dead inputs in the reference.
  float* pn  = (float*)d_ws;                         // 65536 floats = 256 KB
  float* out = (float*)d_out;                        // (4096, 64) fp32

  pnorm_kernel<<<M_PTS / 256, 256, 0, stream>>>(P, pn);
  nn_kernel<<<N_PTS / 16, 256, 0, stream>>>(X, P, pn, out);
}